// ActorGraphPolicy_46402826666062
// MI455X (gfx1250) — compile-verified
//
#include <hip/hip_runtime.h>
#include <math.h>

// ---------------------------------------------------------------------------
// ActorGraphPolicy on MI455X (gfx1250, wave32).
// Down-pass MLP GEMMs (97% of FLOPs) run on V_WMMA_F32_16X16X4_F32 (fp32
// matrix pipe, matches reference numerics). Weights are repacked once per
// launch into a padded fragment-native layout so the GEMM K-loop has zero
// branches and pure b64 loads -> software-pipelinable WMMA stream.
// ---------------------------------------------------------------------------

typedef float v2f __attribute__((ext_vector_type(2)));
typedef float v8f __attribute__((ext_vector_type(8)));

#define NUM_LIMBS 12
#define STATE_DIM 11
#define MSG_DIM   64

// ---------------------------------------------------------------------------
// Weight repack: WP = [ biasPad(Npad) | (K/4) x Npad x 4 ]
//   WP[Npad + (k4*Npad + col)*4 + j] = W[(4*k4+j)*N + col]  (0 if col >= N)
// Lane (half,l16) then fetches its B fragment (elements K=kk,kk+1 of column
// col) as ONE aligned 8-byte load.
// ---------------------------------------------------------------------------
__global__ __launch_bounds__(256) void pack_weights_kernel(
    const float* __restrict__ W, const float* __restrict__ bias,
    float* __restrict__ WP, int K, int N, int Npad)
{
    long total = (long)Npad + (long)(K / 4) * Npad * 4;
    long idx = (long)blockIdx.x * blockDim.x + threadIdx.x;
    if (idx >= total) return;
    if (idx < Npad) {
        WP[idx] = (idx < N) ? bias[idx] : 0.f;
    } else {
        long e = idx - Npad;
        int j   = (int)(e & 3);
        long t  = e >> 2;
        int col = (int)(t % Npad);
        int k4  = (int)(t / Npad);
        int k   = k4 * 4 + j;
        WP[idx] = (col < N) ? W[(long)k * N + col] : 0.f;
    }
}

// ---------------------------------------------------------------------------
// WMMA GEMM: C[M,N] = act(A[M,K] @ W[K,N] + bias)   (W,bias packed in WP)
// K % 4 == 0. Block = 8 waves; wave computes 32(M) x 64(N), looping N chunks.
// Per K-step: 2 A-frag b64 loads + 4 B-frag b64 loads + 8 wmma, double
// buffered so loads for k+4 are in flight during the k multiplies.
// ---------------------------------------------------------------------------
#define WMMA4(a, b, c) __builtin_amdgcn_wmma_f32_16x16x4_f32( \
    false, (a), false, (b), (short)0, (c), false, false)

__global__ __launch_bounds__(256) void gemm_wmma_kernel(
    const float* __restrict__ A, const float* __restrict__ WP,
    float* __restrict__ C, int M, int K, int N, int Npad, int relu)
{
    const int wave = threadIdx.x >> 5;
    const int lane = threadIdx.x & 31;
    const int half = lane >> 4;   // 0: lanes 0-15, 1: lanes 16-31
    const int l16  = lane & 15;
    const long m0 = ((long)blockIdx.x * 8 + wave) * 32;
    if (m0 >= M) return;

    const float* aA = A + (long)(m0 + l16) * K + half * 2;        // rows m0..m0+15
    const float* aB = A + (long)(m0 + 16 + l16) * K + half * 2;   // rows m0+16..m0+31
    const long wstep = (long)Npad * 4;                            // floats per k4 step

    for (int n0 = 0; n0 < N; n0 += 64) {
        v8f acc[2][4];
        #pragma unroll
        for (int t = 0; t < 4; ++t) {
            float bv = WP[n0 + 16 * t + l16];   // padded bias, no bounds check
            v8f ci;
            #pragma unroll
            for (int v = 0; v < 8; ++v) ci[v] = bv;
            acc[0][t] = ci;
            acc[1][t] = ci;
        }

        const float* wq = WP + Npad + (long)(n0 + l16) * 4 + half * 2;

        // prologue: fragments for k = 0
        v2f a0 = *(const v2f*)(aA);
        v2f a1 = *(const v2f*)(aB);
        v2f b0 = *(const v2f*)(wq);
        v2f b1 = *(const v2f*)(wq + 64);
        v2f b2 = *(const v2f*)(wq + 128);
        v2f b3 = *(const v2f*)(wq + 192);

        for (int k = 4; k < K; k += 4) {
            const float* wqk = wq + (long)(k >> 2) * wstep;
            v2f na0 = *(const v2f*)(aA + k);
            v2f na1 = *(const v2f*)(aB + k);
            v2f nb0 = *(const v2f*)(wqk);
            v2f nb1 = *(const v2f*)(wqk + 64);
            v2f nb2 = *(const v2f*)(wqk + 128);
            v2f nb3 = *(const v2f*)(wqk + 192);

            acc[0][0] = WMMA4(a0, b0, acc[0][0]);
            acc[1][0] = WMMA4(a1, b0, acc[1][0]);
            acc[0][1] = WMMA4(a0, b1, acc[0][1]);
            acc[1][1] = WMMA4(a1, b1, acc[1][1]);
            acc[0][2] = WMMA4(a0, b2, acc[0][2]);
            acc[1][2] = WMMA4(a1, b2, acc[1][2]);
            acc[0][3] = WMMA4(a0, b3, acc[0][3]);
            acc[1][3] = WMMA4(a1, b3, acc[1][3]);

            a0 = na0; a1 = na1;
            b0 = nb0; b1 = nb1; b2 = nb2; b3 = nb3;
        }
        // epilogue K-step
        acc[0][0] = WMMA4(a0, b0, acc[0][0]);
        acc[1][0] = WMMA4(a1, b0, acc[1][0]);
        acc[0][1] = WMMA4(a0, b1, acc[0][1]);
        acc[1][1] = WMMA4(a1, b1, acc[1][1]);
        acc[0][2] = WMMA4(a0, b2, acc[0][2]);
        acc[1][2] = WMMA4(a1, b2, acc[1][2]);
        acc[0][3] = WMMA4(a0, b3, acc[0][3]);
        acc[1][3] = WMMA4(a1, b3, acc[1][3]);

        #pragma unroll
        for (int t = 0; t < 4; ++t) {
            int col = n0 + 16 * t + l16;
            if (col < N) {
                #pragma unroll
                for (int mi = 0; mi < 2; ++mi) {
                    #pragma unroll
                    for (int v = 0; v < 8; ++v) {
                        long row = m0 + mi * 16 + v + half * 8;
                        float x = acc[mi][t][v];
                        if (relu) x = fmaxf(x, 0.f);
                        C[row * N + col] = x;
                    }
                }
            }
        }
    }
}

// ---------------------------------------------------------------------------
// Up-pass: one thread per batch row, one launch per node (leaves -> root).
// ---------------------------------------------------------------------------
__global__ __launch_bounds__(256) void up_pass_kernel(
    const float* __restrict__ state,
    const float* __restrict__ fc1w, const float* __restrict__ fc1b,
    const float* __restrict__ attw, const float* __restrict__ attb,
    const float* __restrict__ fc2w, const float* __restrict__ fc2b,
    const float* __restrict__ fc3w, const float* __restrict__ fc3b,
    const float* __restrict__ msg_up_base, float* __restrict__ msg_up_out,
    int node, int c0, int c1, int batch)
{
    int row = blockIdx.x * blockDim.x + threadIdx.x;
    if (row >= batch) return;

    const float* s = state + (long)row * (NUM_LIMBS * STATE_DIM) + node * STATE_DIM;
    float sv[STATE_DIM];
    #pragma unroll
    for (int i = 0; i < STATE_DIM; ++i) sv[i] = s[i];

    float cs[2][20];
    int ch[2] = {c0, c1};
    #pragma unroll
    for (int j = 0; j < 2; ++j) {
        #pragma unroll
        for (int k = 0; k < 20; ++k) cs[j][k] = 0.f;
        if (ch[j] >= 0) {
            const float* c = state + (long)row * (NUM_LIMBS * STATE_DIM) + ch[j] * STATE_DIM;
            float r0 = c[0] - sv[0], r1 = c[1] - sv[1], r2 = c[2] - sv[2];
            cs[j][0] = fabsf(r0); cs[j][1] = fabsf(r1); cs[j][2] = fabsf(r2);
            cs[j][3] = r0;        cs[j][4] = r1;        cs[j][5] = r2;
            cs[j][6] = sv[0];     cs[j][7] = sv[1];     cs[j][8] = sv[2];
            cs[j][9] = c[0];      cs[j][10] = c[1];     cs[j][11] = c[2];
            #pragma unroll
            for (int k = 0; k < 8; ++k) cs[j][12 + k] = c[3 + k];
        }
    }
    const float* mu0 = (c0 >= 0) ? msg_up_base + ((long)c0 * batch + row) * MSG_DIM : nullptr;
    const float* mu1 = (c1 >= 0) ? msg_up_base + ((long)c1 * batch + row) * MSG_DIM : nullptr;

    float xm[128];
    float nrm = 0.f;
    for (int f = 0; f < 64; ++f) {
        float v = fc1b[f];
        #pragma unroll
        for (int i = 0; i < STATE_DIM; ++i) v += sv[i] * fc1w[i * 64 + f];
        xm[f] = v;
        nrm += v * v;
    }
    float sc = 1.f / fmaxf(sqrtf(nrm), 1e-12f);
    for (int f = 0; f < 64; ++f) xm[f] *= sc;

    for (int f = 0; f < 64; ++f) {
        float l0 = attb[f], l1 = attb[f];
        #pragma unroll
        for (int k = 0; k < 20; ++k) {
            float w = attw[k * 64 + f];
            l0 += cs[0][k] * w;
            l1 += cs[1][k] * w;
        }
        float mx = fmaxf(l0, l1);
        float e0 = expf(l0 - mx), e1 = expf(l1 - mx);
        float inv = 1.f / (e0 + e1);
        float m0v = mu0 ? mu0[f] : 0.f;
        float m1v = mu1 ? mu1[f] : 0.f;
        xm[64 + f] = (m0v * e0 + m1v * e1) * inv;
    }

    for (int f = 0; f < 128; ++f) xm[f] = tanhf(xm[f]);

    float h[64];
    for (int f = 0; f < 64; ++f) {
        float v = fc2b[f];
        #pragma unroll
        for (int i = 0; i < 128; ++i) v += xm[i] * fc2w[i * 64 + f];
        h[f] = tanhf(v);
    }

    nrm = 0.f;
    for (int f = 0; f < 64; ++f) {
        float v = fc3b[f];
        #pragma unroll
        for (int i = 0; i < 64; ++i) v += h[i] * fc3w[i * 64 + f];
        xm[f] = v;
        nrm += v * v;
    }
    sc = 1.f / fmaxf(sqrtf(nrm), 1e-12f);
    float* out = msg_up_out + ((long)node * batch + row) * MSG_DIM;
    for (int f = 0; f < 64; ++f) out[f] = xm[f] * sc;
}

// ---------------------------------------------------------------------------
// Build xm[B,140] = tanh(concat(rel(12), msg_up[node](64), msg_in(64)))
// ---------------------------------------------------------------------------
__global__ __launch_bounds__(256) void build_xm_kernel(
    const float* __restrict__ state, const float* __restrict__ msg_up,
    const float* __restrict__ msg_down, float* __restrict__ xm,
    int node, int prel, int pmsg, int cidx, int batch)
{
    long idx = (long)blockIdx.x * blockDim.x + threadIdx.x;
    long total = (long)batch * 140;
    if (idx >= total) return;
    int row = (int)(idx / 140);
    int f   = (int)(idx % 140);
    float v;
    if (f < 12) {
        const float* sp = state + (long)row * (NUM_LIMBS * STATE_DIM);
        int g = f / 3, c = f % 3;
        float np = sp[node * STATE_DIM + c];
        float pp = sp[prel * STATE_DIM + c];   // parent pos (node 11 for root, python -1 idx)
        float rel = np - pp;
        v = (g == 0) ? fabsf(rel) : (g == 1) ? rel : (g == 2) ? pp : np;
    } else if (f < 76) {
        v = msg_up[((long)node * batch + row) * MSG_DIM + (f - 12)];
    } else {
        v = (pmsg < 0) ? 0.f
            : msg_down[((long)pmsg * batch + row) * 128 + cidx * 64 + (f - 76)];
    }
    xm[(long)row * 140 + f] = tanhf(v);
}

// ---------------------------------------------------------------------------
// Action head: out[row,node] = tanh(h2[row,:] @ w[300,1] + b)   (wave per row)
// ---------------------------------------------------------------------------
__global__ __launch_bounds__(256) void act_head_kernel(
    const float* __restrict__ h2, const float* __restrict__ w,
    const float* __restrict__ b, float* __restrict__ out, int node, int batch)
{
    int wave = threadIdx.x >> 5, lane = threadIdx.x & 31;
    long row = (long)blockIdx.x * 8 + wave;
    if (row >= batch) return;
    float sum = 0.f;
    for (int i = lane; i < 300; i += 32) sum += h2[row * 300 + i] * w[i];
    #pragma unroll
    for (int m = 16; m >= 1; m >>= 1) sum += __shfl_xor(sum, m, 32);
    if (lane == 0) out[row * NUM_LIMBS + node] = tanhf(sum + b[0]);
}

// ---------------------------------------------------------------------------
// In-place l2norm over 128 features (wave per row)
// ---------------------------------------------------------------------------
__global__ __launch_bounds__(256) void l2norm128_kernel(float* __restrict__ m, int batch)
{
    int wave = threadIdx.x >> 5, lane = threadIdx.x & 31;
    long row = (long)blockIdx.x * 8 + wave;
    if (row >= batch) return;
    float* p = m + row * 128;
    float v0 = p[lane], v1 = p[lane + 32], v2 = p[lane + 64], v3 = p[lane + 96];
    float s = v0 * v0 + v1 * v1 + v2 * v2 + v3 * v3;
    #pragma unroll
    for (int mk = 16; mk >= 1; mk >>= 1) s += __shfl_xor(s, mk, 32);
    float sc = 1.f / fmaxf(sqrtf(s), 1e-12f);
    p[lane] = v0 * sc; p[lane + 32] = v1 * sc;
    p[lane + 64] = v2 * sc; p[lane + 96] = v3 * sc;
}

// ---------------------------------------------------------------------------
extern "C" void kernel_launch(void* const* d_in, const int* in_sizes, int n_in,
                              void* d_out, int out_size, void* d_ws, size_t ws_size,
                              hipStream_t stream)
{
    const float* state    = (const float*)d_in[0];
    const float* up_fc1_w = (const float*)d_in[1];
    const float* up_fc1_b = (const float*)d_in[2];
    const float* up_fc2_w = (const float*)d_in[3];
    const float* up_fc2_b = (const float*)d_in[4];
    const float* up_fc3_w = (const float*)d_in[5];
    const float* up_fc3_b = (const float*)d_in[6];
    const float* up_att_w = (const float*)d_in[7];
    const float* up_att_b = (const float*)d_in[8];
    const float* act_l1_w = (const float*)d_in[9];
    const float* act_l1_b = (const float*)d_in[10];
    const float* act_l2_w = (const float*)d_in[11];
    const float* act_l2_b = (const float*)d_in[12];
    const float* act_l3_w = (const float*)d_in[13];
    const float* act_l3_b = (const float*)d_in[14];
    const float* msg_l1_w = (const float*)d_in[15];
    const float* msg_l1_b = (const float*)d_in[16];
    const float* msg_l2_w = (const float*)d_in[17];
    const float* msg_l2_b = (const float*)d_in[18];
    const float* msg_l3_w = (const float*)d_in[19];
    const float* msg_l3_b = (const float*)d_in[20];

    const int B = in_sizes[0] / (NUM_LIMBS * STATE_DIM);   // 32768
    (void)ws_size; (void)out_size; (void)n_in;

    // ---- workspace layout (floats) ----
    float* ws = (float*)d_ws;
    size_t off = 0;
    float* msg_up = ws + off;   off += (size_t)NUM_LIMBS * B * 64;
    float* msg_dn = ws + off;   off += (size_t)NUM_LIMBS * B * 128;
    float* xm     = ws + off;   off += (size_t)B * 140;
    float* h1     = ws + off;   off += (size_t)B * 400;
    float* h2     = ws + off;   off += (size_t)B * 300;
    // packed weights: Npad + (K/4)*Npad*4 floats each
    const int NP1 = 448, NP2 = 320, NP3 = 128;   // padded N for l1/l2/msg_l3
    const size_t SZ1 = (size_t)NP1 + (size_t)(140 / 4) * NP1 * 4;
    const size_t SZ2 = (size_t)NP2 + (size_t)(400 / 4) * NP2 * 4;
    const size_t SZ3 = (size_t)NP3 + (size_t)(300 / 4) * NP3 * 4;
    float* act1p = ws + off;    off += SZ1;
    float* act2p = ws + off;    off += SZ2;
    float* msg1p = ws + off;    off += SZ1;
    float* msg2p = ws + off;    off += SZ2;
    float* msg3p = ws + off;    off += SZ3;

    // ---- pack weights (runs every call; ~1.7MB total, negligible) ----
    {
        int blk = 256;
        pack_weights_kernel<<<(int)((SZ1 + blk - 1) / blk), blk, 0, stream>>>(act_l1_w, act_l1_b, act1p, 140, 400, NP1);
        pack_weights_kernel<<<(int)((SZ2 + blk - 1) / blk), blk, 0, stream>>>(act_l2_w, act_l2_b, act2p, 400, 300, NP2);
        pack_weights_kernel<<<(int)((SZ1 + blk - 1) / blk), blk, 0, stream>>>(msg_l1_w, msg_l1_b, msg1p, 140, 400, NP1);
        pack_weights_kernel<<<(int)((SZ2 + blk - 1) / blk), blk, 0, stream>>>(msg_l2_w, msg_l2_b, msg2p, 400, 300, NP2);
        pack_weights_kernel<<<(int)((SZ3 + blk - 1) / blk), blk, 0, stream>>>(msg_l3_w, msg_l3_b, msg3p, 300, 128, NP3);
    }

    // tree topology (PARENTS = [-1,0,0,1,1,2,2,3,4,5,6,7])
    static const int CH0[NUM_LIMBS]  = { 1, 3, 5, 7, 8, 9, 10, 11, -1, -1, -1, -1 };
    static const int CH1[NUM_LIMBS]  = { 2, 4, 6, -1, -1, -1, -1, -1, -1, -1, -1, -1 };
    static const int PAR[NUM_LIMBS]  = { -1, 0, 0, 1, 1, 2, 2, 3, 4, 5, 6, 7 };
    static const int CIDX[NUM_LIMBS] = { 0, 0, 1, 0, 1, 0, 1, 0, 0, 0, 0, 0 };

    // ---------- up pass (leaves -> root) ----------
    for (int node = NUM_LIMBS - 1; node >= 0; --node) {
        up_pass_kernel<<<(B + 255) / 256, 256, 0, stream>>>(
            state, up_fc1_w, up_fc1_b, up_att_w, up_att_b,
            up_fc2_w, up_fc2_b, up_fc3_w, up_fc3_b,
            msg_up, msg_up, node, CH0[node], CH1[node], B);
    }

    // ---------- down pass (root -> leaves) ----------
    const int gemm_blocks = B / 256;          // 32 rows/wave * 8 waves
    const int wave_blocks = B / 8;            // wave-per-row kernels
    for (int node = 0; node < NUM_LIMBS; ++node) {
        int prel = (PAR[node] < 0) ? (NUM_LIMBS - 1) : PAR[node];
        int pmsg = PAR[node];
        long total = (long)B * 140;
        build_xm_kernel<<<(int)((total + 255) / 256), 256, 0, stream>>>(
            state, msg_up, msg_dn, xm, node, prel, pmsg, CIDX[node], B);

        // action branch
        gemm_wmma_kernel<<<gemm_blocks, 256, 0, stream>>>(xm, act1p, h1, B, 140, 400, NP1, 1);
        gemm_wmma_kernel<<<gemm_blocks, 256, 0, stream>>>(h1, act2p, h2, B, 400, 300, NP2, 1);
        act_head_kernel<<<wave_blocks, 256, 0, stream>>>(h2, act_l3_w, act_l3_b, (float*)d_out, node, B);

        // message branch
        gemm_wmma_kernel<<<gemm_blocks, 256, 0, stream>>>(xm, msg1p, h1, B, 140, 400, NP1, 1);
        gemm_wmma_kernel<<<gemm_blocks, 256, 0, stream>>>(h1, msg2p, h2, B, 400, 300, NP2, 1);
        float* md = msg_dn + (size_t)node * B * 128;
        gemm_wmma_kernel<<<gemm_blocks, 256, 0, stream>>>(h2, msg3p, md, B, 300, 128, NP3, 0);
        l2norm128_kernel<<<wave_blocks, 256, 0, stream>>>(md, B);
    }
}